// GaussianKernel_35562329211335
// MI455X (gfx1250) — compile-verified
//
#include <hip/hip_runtime.h>
#include <math.h>

typedef __attribute__((ext_vector_type(2))) float v2f;
typedef __attribute__((ext_vector_type(8))) float v8f;

#define D 64
#define KCHUNKS 16          // 64 / 4 (K per V_WMMA_F32_16X16X4_F32)
#define WAVES 8             // waves per block; each owns an N-slice
#define NEG_BIG (-3.0e38f)

// ---------------------------------------------------------------------------
// Pass 1a: total sample weight -> ws[0]
// ---------------------------------------------------------------------------
__global__ __launch_bounds__(256) void kde_wsum_kernel(const float* __restrict__ sw,
                                                       float* __restrict__ wsum, int n) {
    __shared__ float red[256];
    float s = 0.0f;
    for (int i = threadIdx.x; i < n; i += 256) s += sw[i];
    red[threadIdx.x] = s;
    __syncthreads();
    for (int d = 128; d > 0; d >>= 1) {
        if ((int)threadIdx.x < d) red[threadIdx.x] += red[threadIdx.x + d];
        __syncthreads();
    }
    if (threadIdx.x == 0) *wsum = red[0];
}

// ---------------------------------------------------------------------------
// Pass 1b: r2[n] = ||train[n]||^2 -> ws[8..8+N)
// ---------------------------------------------------------------------------
__global__ __launch_bounds__(256) void kde_r2_kernel(const float* __restrict__ trainX,
                                                     float* __restrict__ r2, int N) {
    const int n = blockIdx.x * 256 + threadIdx.x;
    if (n >= N) return;
    const float4* row = (const float4*)(trainX + (size_t)n * D);
    float s = 0.0f;
#pragma unroll
    for (int i = 0; i < D / 4; ++i) {
        float4 q = row[i];
        s += q.x * q.x + q.y * q.y + q.z * q.z + q.w * q.w;
    }
    r2[n] = s;
}

// ---------------------------------------------------------------------------
// Online logsumexp accumulator update (branch avoids the rescale exp in the
// common "no new max" case).
// ---------------------------------------------------------------------------
__device__ __forceinline__ void lse_update(float& m, float& s, float val, float w) {
    if (val <= m) {
        s += w * __expf(val - m);
    } else {
        s = s * __expf(m - val) + w;
        m = val;
    }
}

// ---------------------------------------------------------------------------
// Pass 2: fused cross-products + weighted online logsumexp.
// Block = 16 test rows; 8 waves each cover N/8 train rows (2 tiles/iter);
// merge across waves in LDS.
// ---------------------------------------------------------------------------
template <bool R2PRE>
__global__ __launch_bounds__(256) void kde_main_kernel(
    const float* __restrict__ testX,   // [M, 64]
    const float* __restrict__ trainX,  // [N, 64]
    const float* __restrict__ sw,      // [N] (unnormalized)
    const float* __restrict__ wsum,    // [1]
    const float* __restrict__ r2buf,   // [N] (only if R2PRE)
    float* __restrict__ out,           // [M]
    int N) {
    const int lane = threadIdx.x & 31;
    const int wave = threadIdx.x >> 5;
    const int hi   = lane >> 4;        // half-wave select
    const int lo   = lane & 15;
    const int m0   = blockIdx.x * 16;

    // --- A tile in WMMA A-layout: lanes 0-15 row M=lo cols {4c,4c+1};
    //     lanes 16-31 same rows cols {4c+2,4c+3}
    v2f a[KCHUNKS];
    float t2h = 0.0f;
    const float* arow = testX + (size_t)(m0 + lo) * D + 2 * hi;
#pragma unroll
    for (int c = 0; c < KCHUNKS; ++c) {
        a[c] = *(const v2f*)(arow + 4 * c);
        t2h += a[c].x * a[c].x + a[c].y * a[c].y;
    }
    const float Z = 32.0f * 1.8378770664093455f;   // 0.5*64*log(2*pi), h=1
    float t2 = t2h + __shfl_xor(t2h, 16);          // ||test[m0+lo]||^2, both halves
    // redistribute into C-matrix order: accumulator v <-> row (v + 8*hi)
    float cv[8];
#pragma unroll
    for (int v = 0; v < 8; ++v) cv[v] = -0.5f * __shfl(t2, v + 8 * hi) - Z;

    float rm[8], rs[8];
#pragma unroll
    for (int v = 0; v < 8; ++v) { rm[v] = NEG_BIG; rs[v] = 0.0f; }

    const int nSlice = N / WAVES;                   // divisible by 32
    const int nBeg = wave * nSlice;
    const int nEnd = nBeg + nSlice;
    const float* bbase = trainX + 2 * hi;

    for (int n0 = nBeg; n0 < nEnd; n0 += 32) {
        const int na = n0 + lo;
        const int nb = na + 16;
        const float* browA = bbase + (size_t)na * D;
        const float* browB = bbase + (size_t)nb * D;
        const float swA = sw[na];
        const float swB = sw[nb];
        float baseA = 0.0f, baseB = 0.0f;
        if (R2PRE) {
            baseA = -0.5f * r2buf[na];
            baseB = -0.5f * r2buf[nb];
        }

        v8f acc0 = {};
        v8f acc1 = {};
        float r2hA = 0.0f, r2hB = 0.0f;
#pragma unroll
        for (int c = 0; c < KCHUNKS; ++c) {
            v2f b0 = *(const v2f*)(browA + 4 * c);
            v2f b1 = *(const v2f*)(browB + 4 * c);
            if (!R2PRE) {
                r2hA += b0.x * b0.x + b0.y * b0.y;
                r2hB += b1.x * b1.x + b1.y * b1.y;
            }
            acc0 = __builtin_amdgcn_wmma_f32_16x16x4_f32(
                false, a[c], false, b0, (short)0, acc0, false, false);
            acc1 = __builtin_amdgcn_wmma_f32_16x16x4_f32(
                false, a[c], false, b1, (short)0, acc1, false, false);
        }
        if (!R2PRE) {
            baseA = -0.5f * (r2hA + __shfl_xor(r2hA, 16));
            baseB = -0.5f * (r2hB + __shfl_xor(r2hB, 16));
        }

#pragma unroll
        for (int v = 0; v < 8; ++v) {
            lse_update(rm[v], rs[v], (acc0[v] + cv[v]) + baseA, swA);
            lse_update(rm[v], rs[v], (acc1[v] + cv[v]) + baseB, swB);
        }
    }

    // --- intra-wave merge across the 16 N-lanes of each half-wave
    __shared__ float lds_m[WAVES][16];
    __shared__ float lds_s[WAVES][16];
#pragma unroll
    for (int v = 0; v < 8; ++v) {
        float m = rm[v], s = rs[v];
#pragma unroll
        for (int d = 1; d < 16; d <<= 1) {
            float om = __shfl_xor(m, d);
            float os = __shfl_xor(s, d);
            float nm = fmaxf(m, om);
            s = s * __expf(m - nm) + os * __expf(om - nm);
            m = nm;
        }
        if (lo == 0) {                              // lane 0 -> row v, lane 16 -> row v+8
            lds_m[wave][v + 8 * hi] = m;
            lds_s[wave][v + 8 * hi] = s;
        }
    }
    __syncthreads();

    // --- cross-wave merge + output (threads 0..15)
    if (threadIdx.x < 16) {
        const int row = threadIdx.x;
        float m = lds_m[0][row], s = lds_s[0][row];
#pragma unroll
        for (int w = 1; w < WAVES; ++w) {
            float om = lds_m[w][row];
            float os = lds_s[w][row];
            float nm = fmaxf(m, om);
            s = s * __expf(m - nm) + os * __expf(om - nm);
            m = nm;
        }
        const float W = *wsum;
        out[m0 + row] = m + __logf(s / W + 1e-20f);
    }
}

// ---------------------------------------------------------------------------
extern "C" void kernel_launch(void* const* d_in, const int* in_sizes, int n_in,
                              void* d_out, int out_size, void* d_ws, size_t ws_size,
                              hipStream_t stream) {
    const float* testX  = (const float*)d_in[0];   // [8192, 64]
    const float* trainX = (const float*)d_in[1];   // [16384, 64]
    const float* sw     = (const float*)d_in[2];   // [16384]
    float* out  = (float*)d_out;                   // [8192]
    float* wsum = (float*)d_ws;                    // ws[0]
    float* r2buf = (float*)d_ws + 8;               // ws[8..8+N)

    const int M = in_sizes[0] / D;
    const int N = in_sizes[2];

    hipLaunchKernelGGL(kde_wsum_kernel, dim3(1), dim3(256), 0, stream, sw, wsum, N);

    const bool r2pre = ws_size >= (size_t)(8 + N) * sizeof(float);
    if (r2pre) {
        hipLaunchKernelGGL(kde_r2_kernel, dim3((N + 255) / 256), dim3(256), 0, stream,
                           trainX, r2buf, N);
        hipLaunchKernelGGL((kde_main_kernel<true>), dim3(M / 16), dim3(256), 0, stream,
                           testX, trainX, sw, wsum, r2buf, out, N);
    } else {
        hipLaunchKernelGGL((kde_main_kernel<false>), dim3(M / 16), dim3(256), 0, stream,
                           testX, trainX, sw, wsum, (const float*)nullptr, out, N);
    }
}